// MiniLLM_79388175499590
// MI455X (gfx1250) — compile-verified
//
#include <hip/hip_runtime.h>
#include <math.h>

// MiniLLM last-token forward, MI455X (gfx1250, wave32).
// Only the last token's hidden state reaches the output -> compute K/V for all
// positions, one query per (b,h), 8 softmaxes of length S, and a tiny tail.
// K/V projection uses V_WMMA_F32_16X16X4_F32 (two chained K=4 ops = K dim 8),
// N=16 packs [Wk | Wv] columns, M=16 sequence positions per wave.
// Trig/exp use the hardware TRANS32 ops (v_sin/v_cos/v_exp/v_rsq) instead of
// libm's Payne-Hanek path -- kernel is launch-latency bound, so instruction
// count and branch-free WMMA regions are what matter.

typedef float v2f __attribute__((ext_vector_type(2)));
typedef float v8f __attribute__((ext_vector_type(8)));

#define Bc 4
#define Sc 2048
#define Dc 8
#define Hc 2
#define HDc 4
#define Fc 24
#define Vc 20
#define EPSc 1e-6f
// rope freqs for HD=4: f0 = 1.0, f1 = 10000^-0.5 = 0.01
#define ROPE_F1 0.01f

__device__ __forceinline__ float fast_rsqrt(float v) {
    return __builtin_amdgcn_rsqf(v);           // v_rsq_f32
}

__global__ __launch_bounds__(256) void minillm_proj_score(
    const int* __restrict__ x, const float* __restrict__ emb,
    const float* __restrict__ g1, const float* __restrict__ Wq,
    const float* __restrict__ Wk, const float* __restrict__ Wv,
    float* __restrict__ scores, float* __restrict__ vbuf)
{
    __shared__ float lds[8][16][16];   // [wave][position-in-tile][16 proj dims: k0..7,v0..7]

    const int tid  = threadIdx.x;
    const int wave = tid >> 5;
    const int lane = tid & 31;
    const int col  = lane & 15;        // D-matrix column / A-matrix row (position)
    const int hi   = lane >> 4;        // lane-half selector
    const int tile = blockIdx.x * 8 + wave;   // 512 tiles of 16 positions
    const int b    = tile >> 7;               // 128 tiles per batch row
    const int s0   = (tile & 127) << 4;
    const int p    = s0 + col;                // this lane's position

    // ---- n = rmsnorm(emb[x[b,p]], g1), duplicated across the lane pair ----
    float n0,n1,n2,n3,n4,n5,n6,n7;
    {
        const int t = x[b*Sc + p];
        const float* e = emb + t*Dc;
        n0=e[0]; n1=e[1]; n2=e[2]; n3=e[3]; n4=e[4]; n5=e[5]; n6=e[6]; n7=e[7];
        float ss = n0*n0+n1*n1+n2*n2+n3*n3+n4*n4+n5*n5+n6*n6+n7*n7;
        float inv = fast_rsqrt(ss*0.125f + EPSc);
        n0*=inv*g1[0]; n1*=inv*g1[1]; n2*=inv*g1[2]; n3*=inv*g1[3];
        n4*=inv*g1[4]; n5*=inv*g1[5]; n6*=inv*g1[6]; n7*=inv*g1[7];
    }

    // ---- A matrices (16x4 f32): lanes 0-15 hold K=0,1; lanes 16-31 K=2,3 ----
    v2f a0, a1;
    a0.x = hi ? n2 : n0;  a0.y = hi ? n3 : n1;   // K dims 0..3
    a1.x = hi ? n6 : n4;  a1.y = hi ? n7 : n5;   // K dims 4..7

    // ---- B matrices (4x16 f32): column = col; cols 0..7 = Wk, 8..15 = Wv ----
    // lanes 0-15 hold K rows {0,1} (vgpr0,vgpr1); lanes 16-31 hold {2,3}
    const float* wkv = (col < 8) ? Wk : Wv;
    const int cc = col & 7;
    v2f b0, b1;
    {
        const int k0 = 2*hi;
        b0.x = wkv[(k0+0)*Dc + cc];  b0.y = wkv[(k0+1)*Dc + cc];
        b1.x = wkv[(k0+4)*Dc + cc];  b1.y = wkv[(k0+5)*Dc + cc];
    }

    // ---- D = n(16x8) @ [Wk|Wv](8x16) via two chained K=4 f32 WMMAs ----
    v8f c = {0.f,0.f,0.f,0.f,0.f,0.f,0.f,0.f};
    c = __builtin_amdgcn_wmma_f32_16x16x4_f32(false, a0, false, b0, (short)0, c, false, false);
    c = __builtin_amdgcn_wmma_f32_16x16x4_f32(false, a1, false, b1, (short)0, c, false, false);

    // D layout: lane holds column n=col, rows m = hi*8 + j in c[j]. Transpose via LDS.
    #pragma unroll
    for (int j = 0; j < 8; ++j)
        lds[wave][hi*8 + j][col] = c[j];
    __syncthreads();

    // ---- per-lane q for (b, head=hi), roped at P = S-1 (recomputed, cheap) ----
    float qh0,qh1,qh2,qh3;
    {
        const int t = x[b*Sc + Sc-1];
        const float* e = emb + t*Dc;
        float m0=e[0],m1=e[1],m2=e[2],m3=e[3],m4=e[4],m5=e[5],m6=e[6],m7=e[7];
        float ss = m0*m0+m1*m1+m2*m2+m3*m3+m4*m4+m5*m5+m6*m6+m7*m7;
        float inv = fast_rsqrt(ss*0.125f + EPSc);
        m0*=inv*g1[0]; m1*=inv*g1[1]; m2*=inv*g1[2]; m3*=inv*g1[3];
        m4*=inv*g1[4]; m5*=inv*g1[5]; m6*=inv*g1[6]; m7*=inv*g1[7];
        const int j0 = 4*hi;                       // this head's 4 output dims
        float q[4];
        #pragma unroll
        for (int jj = 0; jj < 4; ++jj) {
            const float* wq = Wq + (j0 + jj);
            q[jj] = m0*wq[0] + m1*wq[Dc] + m2*wq[2*Dc] + m3*wq[3*Dc]
                  + m4*wq[4*Dc] + m5*wq[5*Dc] + m6*wq[6*Dc] + m7*wq[7*Dc];
        }
        const float P = (float)(Sc - 1);
        float ca = __cosf(P),         sa = __sinf(P);          // v_cos/v_sin
        float cb = __cosf(P*ROPE_F1), sb = __sinf(P*ROPE_F1);
        qh0 = q[0]*ca - q[1]*sa;  qh1 = q[0]*sa + q[1]*ca;
        qh2 = q[2]*cb - q[3]*sb;  qh3 = q[2]*sb + q[3]*cb;
    }

    // ---- read back this position's k/v slice for head hi, rope k, score ----
    const int d0 = 4*hi;
    float k0v = lds[wave][col][d0+0], k1v = lds[wave][col][d0+1];
    float k2v = lds[wave][col][d0+2], k3v = lds[wave][col][d0+3];
    float v0v = lds[wave][col][8+d0+0], v1v = lds[wave][col][8+d0+1];
    float v2v = lds[wave][col][8+d0+2], v3v = lds[wave][col][8+d0+3];

    const float fp = (float)p;
    float ca = __cosf(fp),         sa = __sinf(fp);
    float cb = __cosf(fp*ROPE_F1), sb = __sinf(fp*ROPE_F1);
    float rk0 = k0v*ca - k1v*sa, rk1 = k0v*sa + k1v*ca;
    float rk2 = k2v*cb - k3v*sb, rk3 = k2v*sb + k3v*cb;

    const float sc = 0.5f * (qh0*rk0 + qh1*rk1 + qh2*rk2 + qh3*rk3); // 1/sqrt(HD)=0.5
    scores[(b*Hc + hi)*Sc + p] = sc;

    float* vp = vbuf + (b*Sc + p)*Dc + d0;
    vp[0]=v0v; vp[1]=v1v; vp[2]=v2v; vp[3]=v3v;
}

__global__ __launch_bounds__(256) void minillm_attn_tail(
    const int* __restrict__ x, const float* __restrict__ emb,
    const float* __restrict__ Wo, const float* __restrict__ g2,
    const float* __restrict__ Wup, const float* __restrict__ Wdn,
    const float* __restrict__ gf, const float* __restrict__ Wvb,
    const float* __restrict__ scores, const float* __restrict__ vbuf,
    float* __restrict__ out)
{
    __shared__ float red[256];
    __shared__ float racc[4][256];
    __shared__ float ctx8[8];

    const int b = blockIdx.x;
    const int tid = threadIdx.x;

    for (int h = 0; h < Hc; ++h) {
        const float* sc = scores + (b*Hc + h)*Sc;
        // pass A: max
        float lm = -INFINITY;
        for (int s = tid; s < Sc; s += 256) lm = fmaxf(lm, sc[s]);
        red[tid] = lm; __syncthreads();
        for (int off = 128; off > 0; off >>= 1) {
            if (tid < off) red[tid] = fmaxf(red[tid], red[tid+off]);
            __syncthreads();
        }
        const float M = red[0]; __syncthreads();
        // pass B: exp-sum + weighted V
        float sum = 0.f, a0 = 0.f, a1 = 0.f, a2 = 0.f, a3 = 0.f;
        for (int s = tid; s < Sc; s += 256) {
            float e = __expf(sc[s] - M);               // v_exp_f32
            const float* vp = vbuf + (b*Sc + s)*Dc + 4*h;
            sum += e; a0 += e*vp[0]; a1 += e*vp[1]; a2 += e*vp[2]; a3 += e*vp[3];
        }
        red[tid]=sum; racc[0][tid]=a0; racc[1][tid]=a1; racc[2][tid]=a2; racc[3][tid]=a3;
        __syncthreads();
        for (int off = 128; off > 0; off >>= 1) {
            if (tid < off) {
                red[tid] += red[tid+off];
                racc[0][tid] += racc[0][tid+off];
                racc[1][tid] += racc[1][tid+off];
                racc[2][tid] += racc[2][tid+off];
                racc[3][tid] += racc[3][tid+off];
            }
            __syncthreads();
        }
        if (tid == 0) {
            float inv = 1.f / red[0];
            ctx8[4*h+0]=racc[0][0]*inv; ctx8[4*h+1]=racc[1][0]*inv;
            ctx8[4*h+2]=racc[2][0]*inv; ctx8[4*h+3]=racc[3][0]*inv;
        }
        __syncthreads();
    }

    if (tid == 0) {
        // residual: h = emb_last + ctx @ Wo
        const int t = x[b*Sc + Sc-1];
        float h8[Dc];
        #pragma unroll
        for (int i = 0; i < Dc; ++i) {
            float acc = 0.f;
            #pragma unroll
            for (int j = 0; j < Dc; ++j) acc += ctx8[j] * Wo[j*Dc + i];
            h8[i] = emb[t*Dc + i] + acc;
        }
        // MLP: h += relu(rmsnorm(h,g2) @ Wup) @ Wdown
        float ss = 0.f;
        #pragma unroll
        for (int i = 0; i < Dc; ++i) ss += h8[i]*h8[i];
        float inv = fast_rsqrt(ss*0.125f + EPSc);
        float u[Dc];
        #pragma unroll
        for (int i = 0; i < Dc; ++i) u[i] = h8[i]*inv*g2[i];
        for (int f = 0; f < Fc; ++f) {
            float acc = 0.f;
            #pragma unroll
            for (int i = 0; i < Dc; ++i) acc += u[i] * Wup[i*Fc + f];
            acc = fmaxf(acc, 0.f);
            #pragma unroll
            for (int i = 0; i < Dc; ++i) h8[i] += acc * Wdn[f*Dc + i];
        }
        // final norm + vocab
        float ss2 = 0.f;
        #pragma unroll
        for (int i = 0; i < Dc; ++i) ss2 += h8[i]*h8[i];
        float inv2 = fast_rsqrt(ss2*0.125f + EPSc);
        float fo[Dc];
        #pragma unroll
        for (int i = 0; i < Dc; ++i) fo[i] = h8[i]*inv2*gf[i];
        for (int j = 0; j < Vc; ++j) {
            float acc = 0.f;
            #pragma unroll
            for (int i = 0; i < Dc; ++i) acc += fo[i] * Wvb[i*Vc + j];
            out[b*Vc + j] = acc;
        }
    }
}

extern "C" void kernel_launch(void* const* d_in, const int* in_sizes, int n_in,
                              void* d_out, int out_size, void* d_ws, size_t ws_size,
                              hipStream_t stream)
{
    const int*   x   = (const int*)  d_in[0];
    const float* emb = (const float*)d_in[1];
    const float* g1  = (const float*)d_in[2];
    const float* Wq  = (const float*)d_in[3];
    const float* Wk  = (const float*)d_in[4];
    const float* Wv  = (const float*)d_in[5];
    const float* Wo  = (const float*)d_in[6];
    const float* g2  = (const float*)d_in[7];
    const float* Wup = (const float*)d_in[8];
    const float* Wdn = (const float*)d_in[9];
    const float* gf  = (const float*)d_in[10];
    const float* Wvb = (const float*)d_in[11];

    float* scores = (float*)d_ws;              // B*H*S   = 16384 floats
    float* vbuf   = scores + Bc*Hc*Sc;         // B*S*D   = 65536 floats

    // 512 wave-tiles of 16 positions, 8 waves per 256-thread block
    minillm_proj_score<<<64, 256, 0, stream>>>(x, emb, g1, Wq, Wk, Wv, scores, vbuf);
    // one block per batch row: softmax over 2048 keys x 2 heads + scalar tail
    minillm_attn_tail<<<Bc, 256, 0, stream>>>(x, emb, Wo, g2, Wup, Wdn, gf, Wvb,
                                              scores, vbuf, (float*)d_out);
}